// ParametricContinuousConv_79517024518540
// MI455X (gfx1250) — compile-verified
//
#include <hip/hip_runtime.h>

// ---------------------------------------------------------------------------
// ParametricContinuousConv for MI455X (gfx1250, wave32).
// Bandwidth-bound problem (80MB output zero+scatter dominates; ~2 GFLOP of
// matmul) -> keep full f32 precision and use V_WMMA_F32_16X16X4_F32 for the
// two GEMMs (hidden@W2, msg@conv_w^T).
// Register-array indices are always compile-time constants; lane-dependent
// values (off = hi?2:0) appear only in memory addresses, never in register
// indexing (avoids v_cndmask select chains).
// ---------------------------------------------------------------------------

#define C_DIM 64
#define H_DIM 256
#define W_DIM 1216
#define HW    (H_DIM * W_DIM)     // 311296
#define NPTS  40000
#define KNN   9
#define CHID  32
#define BN_EPS 1e-5f

typedef __attribute__((ext_vector_type(2))) float v2f;
typedef __attribute__((ext_vector_type(8))) float v8f;

// D = A(16x4,f32) * B(4x16,f32) + C(16x16,f32)
// args: (neg_a, A, neg_b, B, c_mod, C, reuse_a, reuse_b)
__device__ __forceinline__ v8f wmma4(v2f a, v2f b, v8f c) {
  return __builtin_amdgcn_wmma_f32_16x16x4_f32(
      false, a, false, b, (short)0, c, false, false);
}

// ---------------------------------------------------------------------------
// Kernel 1: zero dense output [64,256,1216] with 128-bit stores.
// ---------------------------------------------------------------------------
__global__ __launch_bounds__(256) void k_zero(float4* __restrict__ out, long n4) {
  long i = (long)blockIdx.x * blockDim.x + threadIdx.x;
  if (i < n4) out[i] = make_float4(0.f, 0.f, 0.f, 0.f);
}

// Kernel 1b: zero BN accumulators (sum[64], sumsq[64]) every call.
__global__ void k_zero_stats(float* __restrict__ s) {
  if (threadIdx.x < 2 * C_DIM) s[threadIdx.x] = 0.f;
}

// ---------------------------------------------------------------------------
// Kernel 2: per point n (one wave32 each):
//   hidden = relu(diff @ W1 + b1)                  (VALU, K=3 too small)
//   wts    = relu(hidden @ W2 + b2)                (WMMA f32 16x16x4, K=32)
//   msg[n] = sum_k wts[k][:] * feat[:, iy_k, ix_k]
// A-fragment (16x4 f32): lane L<16 holds row L {K=0,K=1}; lane L+16 {K=2,K=3}.
// Each lane therefore needs only hidden K with K%4 in {off, off+1}; we compute
// exactly those 16 values into ha[kk]/hb[kk] (constant-indexed).
// C-fragment: VGPR j = rows {j (lanes 0-15), j+8 (lanes 16-31)}, col = lane&15.
// Rows 9..15 of the M=16 tile are padding; excluded from the K-reduce.
// ---------------------------------------------------------------------------
__global__ __launch_bounds__(256) void k_mlp_gather(
    const float* __restrict__ feat, const float* __restrict__ diff,
    const int* __restrict__ nnpix,
    const float* __restrict__ W1, const float* __restrict__ b1,
    const float* __restrict__ W2, const float* __restrict__ b2,
    float* __restrict__ msg) {
  const int wave = threadIdx.x >> 5;
  const int lane = threadIdx.x & 31;
  const int n = blockIdx.x * 8 + wave;     // uniform per wave
  if (n >= NPTS) return;                   // whole wave exits together

  const int row = lane & 15;
  const bool hi = lane >= 16;
  const int off = hi ? 2 : 0;              // used ONLY in memory addresses
  const bool valid = row < KNN;

  // layer 1: lane computes hidden[row][K] only for its own A-fragment slots.
  float d0 = 0.f, d1 = 0.f, d2 = 0.f;
  if (valid) {
    const float* dp = diff + ((long)n * KNN + row) * 3;
    d0 = dp[0]; d1 = dp[1]; d2 = dp[2];
  }
  float ha[8], hb[8];                      // constant-indexed register arrays
#pragma unroll
  for (int kk = 0; kk < 8; ++kk) {
    const int h = 4 * kk + off;            // lane-dependent LOAD address (ok)
    float v0 = fmaf(d0, W1[h],     fmaf(d1, W1[CHID + h],     fmaf(d2, W1[2 * CHID + h],     b1[h])));
    float v1 = fmaf(d0, W1[h + 1], fmaf(d1, W1[CHID + h + 1], fmaf(d2, W1[2 * CHID + h + 1], b1[h + 1])));
    ha[kk] = valid ? fmaxf(v0, 0.f) : 0.f;
    hb[kk] = valid ? fmaxf(v1, 0.f) : 0.f;
  }

  // neighbor pixel offsets (uniform across the wave, L2-resident window)
  int pofs[KNN];
#pragma unroll
  for (int k = 0; k < KNN; ++k) {
    const int ix = nnpix[((long)n * KNN + k) * 2 + 0];
    const int iy = nnpix[((long)n * KNN + k) * 2 + 1];
    pofs[k] = iy * W_DIM + ix;
  }

#pragma unroll
  for (int t = 0; t < 4; ++t) {            // 4 N-tiles of 16 output channels
    v8f acc = {};
#pragma unroll
    for (int kk = 0; kk < 8; ++kk) {       // K=32 in steps of 4
      v2f a, b;
      a.x = ha[kk];
      a.y = hb[kk];
      const float* w2p = W2 + (4 * kk + off) * C_DIM + 16 * t + row;  // B[K][N]
      b.x = w2p[0];
      b.y = w2p[C_DIM];
      acc = wmma4(a, b, acc);
    }
    const int col = 16 * t + row;
    const float b2c = b2[col];
    const float* fc = feat + (long)col * HW;
    float partial = 0.f;
    if (!hi) {                             // rows k = 0..7
#pragma unroll
      for (int j = 0; j < 8; ++j) {
        const float w = fmaxf(acc[j] + b2c, 0.f);
        partial = fmaf(w, fc[pofs[j]], partial);
      }
    } else {                               // row k = 8 lives in VGPR0, hi half
      const float w = fmaxf(acc[0] + b2c, 0.f);
      partial = w * fc[pofs[8]];
    }
    const float other = __shfl_xor(partial, 16, 32);
    if (!hi) msg[(long)n * C_DIM + col] = partial + other;
  }
}

// ---------------------------------------------------------------------------
// Kernel 3: y = msg @ conv_w^T + conv_b via WMMA (M=16 tile/wave, K=64),
// plus per-channel sum/sumsq for BN via LDS float atomics -> global atomics.
// ---------------------------------------------------------------------------
__global__ __launch_bounds__(256) void k_conv_stats(
    const float* __restrict__ msg, const float* __restrict__ convw,
    const float* __restrict__ convb, float* __restrict__ y,
    float* __restrict__ gstats) {
  __shared__ float lsum[2 * C_DIM];
  const int tid = threadIdx.x;
  if (tid < 2 * C_DIM) lsum[tid] = 0.f;
  __syncthreads();

  const int wave = tid >> 5;
  const int lane = tid & 31;
  const int tile = blockIdx.x * 8 + wave;  // uniform per wave
  const int base = tile * 16;
  if (base < NPTS) {
    const int row = lane & 15;
    const bool hi = lane >= 16;
    const int off = hi ? 2 : 0;            // used ONLY in memory addresses
#pragma unroll
    for (int t = 0; t < 4; ++t) {
      v8f acc = {};
#pragma unroll
      for (int kk = 0; kk < 16; ++kk) {    // K=64 in steps of 4
        v2f a, b;
        const float* ap = msg + (long)(base + row) * C_DIM + 4 * kk + off;
        a.x = ap[0]; a.y = ap[1];
        // B[K][N] = conv_w[N][K] (conv_w is [out][in] row-major)
        const float* bp = convw + (long)(16 * t + row) * C_DIM + 4 * kk + off;
        b.x = bp[0]; b.y = bp[1];
        acc = wmma4(a, b, acc);
      }
      const int col = 16 * t + row;
      const float cb = convb[col];
      float s = 0.f, ss = 0.f;
#pragma unroll
      for (int j = 0; j < 8; ++j) {
        const float v = acc[j] + cb;
        const int m = j + (hi ? 8 : 0);    // store ADDRESS only
        y[(long)(base + m) * C_DIM + col] = v;
        s += v;
        ss = fmaf(v, v, ss);
      }
      atomicAdd(&lsum[col], s);            // ds_add_f32
      atomicAdd(&lsum[C_DIM + col], ss);
    }
  }
  __syncthreads();
  if (tid < 2 * C_DIM) atomicAdd(&gstats[tid], lsum[tid]);
}

// ---------------------------------------------------------------------------
// Kernel 4: BatchNorm (training-mode stats over N) + relu + scatter to dense.
// ---------------------------------------------------------------------------
__global__ __launch_bounds__(256) void k_bn_scatter(
    const float* __restrict__ y, const float* __restrict__ gstats,
    const float* __restrict__ gamma, const float* __restrict__ beta,
    const int* __restrict__ pix, float* __restrict__ out) {
  const long i = (long)blockIdx.x * blockDim.x + threadIdx.x;
  if (i >= (long)NPTS * C_DIM) return;
  const int c = (int)(i & (C_DIM - 1));
  const int n = (int)(i >> 6);
  const float inv_n = 1.0f / (float)NPTS;
  const float mean = gstats[c] * inv_n;
  const float var = gstats[C_DIM + c] * inv_n - mean * mean;
  const float norm = (y[i] - mean) * rsqrtf(var + BN_EPS);
  const float r = fmaxf(fmaf(gamma[c], norm, beta[c]), 0.f);
  const int px = pix[2 * n + 0];
  const int py = pix[2 * n + 1];
  out[(long)c * HW + (long)py * W_DIM + px] = r;
}

// ---------------------------------------------------------------------------
extern "C" void kernel_launch(void* const* d_in, const int* in_sizes, int n_in,
                              void* d_out, int out_size, void* d_ws, size_t ws_size,
                              hipStream_t stream) {
  const float* feat  = (const float*)d_in[0];   // [1,64,256,1216]
  const float* diff  = (const float*)d_in[1];   // [1,40000,9,3]
  const int*   pix   = (const int*)d_in[2];     // [1,40000,2]
  const int*   nnpix = (const int*)d_in[3];     // [1,40000,9,2]
  const float* W1    = (const float*)d_in[4];   // [3,32]
  const float* b1    = (const float*)d_in[5];   // [32]
  const float* W2    = (const float*)d_in[6];   // [32,64]
  const float* b2    = (const float*)d_in[7];   // [64]
  const float* convw = (const float*)d_in[8];   // [64,64]
  const float* convb = (const float*)d_in[9];   // [64]
  const float* gamma = (const float*)d_in[10];  // [64]
  const float* beta  = (const float*)d_in[11];  // [64]
  float* out = (float*)d_out;                   // [1,64,256,1216]

  // workspace layout
  float* msg    = (float*)d_ws;                       // 40000*64 f32
  float* yb     = msg + (size_t)NPTS * C_DIM;         // 40000*64 f32
  float* gstats = yb + (size_t)NPTS * C_DIM;          // 128 f32

  const long n4 = (long)C_DIM * HW / 4;
  k_zero<<<(int)((n4 + 255) / 256), 256, 0, stream>>>((float4*)out, n4);
  k_zero_stats<<<1, 128, 0, stream>>>(gstats);
  k_mlp_gather<<<(NPTS + 7) / 8, 256, 0, stream>>>(feat, diff, nnpix, W1, b1, W2, b2, msg);
  k_conv_stats<<<(NPTS / 16 + 7) / 8, 256, 0, stream>>>(msg, convw, convb, yb, gstats);
  const long tot = (long)NPTS * C_DIM;
  k_bn_scatter<<<(int)((tot + 255) / 256), 256, 0, stream>>>(yb, gstats, gamma, beta, pix, out);
}